// GraphAtt_ScaledDotProduct_70471823392983
// MI455X (gfx1250) — compile-verified
//
#include <hip/hip_runtime.h>
#include <hip/hip_bf16.h>
#include <math.h>

// CDNA5 / gfx1250. wave32. Fused masked-attention kernel using
// V_WMMA_F32_16X16X4_F32 (full fp32 accuracy; kernel is HBM-bound on the
// 128MB attn write, so fp32 matrix ops are the right precision here).

typedef __attribute__((ext_vector_type(2))) float v2f;
typedef __attribute__((ext_vector_type(8))) float v8f;

#define N_SEQ   1024
#define DHEAD   64
#define BLKROWS 32          // rows of S per workgroup
#define SROW    1028        // padded LDS stride for the 32x1024 S/P block
#define QROW    66          // padded LDS stride for the 32x64 Q block
#define NEGVAL  (-9.0e15f)

__global__ void __launch_bounds__(256)
GraphAtt_ScaledDotProduct_70471823392983_kernel(
    const float* __restrict__ q, const float* __restrict__ k,
    const float* __restrict__ v, const int* __restrict__ adj,
    float* __restrict__ out, float* __restrict__ attn)
{
    extern __shared__ float lds[];
    float* sP = lds;                       // [32][SROW]  S -> P block
    float* sQ = lds + BLKROWS * SROW;      // [32][QROW]  Q block

    const int rb = blockIdx.x;             // row block (0..31)
    const int ab = blockIdx.y;             // flat batch (0..31), head-major
    const int bb = ab & 3;                 // adj batch = ab % 4
    const int row0 = rb * BLKROWS;

    const float* qb = q   + (size_t)ab * N_SEQ * DHEAD;
    const float* kb = k   + (size_t)ab * N_SEQ * DHEAD;
    const float* vb = v   + (size_t)ab * N_SEQ * DHEAD;
    const int* adjb = adj + (size_t)bb * N_SEQ * N_SEQ;
    float* outb  = out  + (size_t)ab * N_SEQ * DHEAD;
    float* attnb = attn + (size_t)ab * N_SEQ * N_SEQ;

    const int tid = threadIdx.x;
    const int w   = tid >> 5;              // wave id 0..7
    const int l   = tid & 31;              // lane id
    const int ln  = l & 15;                // N / M index within tile
    const int koff = (l < 16) ? 0 : 2;     // half-lane K split (f32 WMMA layout)

    // ---- Stage Q block (32 x 64) into LDS, padded stride 66 ----
    for (int i = tid; i < BLKROWS * (DHEAD / 2); i += 256) {
        int r = i >> 5;                    // row 0..31
        int c = (i & 31) * 2;              // col 0..62 step 2
        v2f t = *(const v2f*)&qb[(size_t)(row0 + r) * DHEAD + c];
        sQ[r * QROW + c]     = t.x;
        sQ[r * QROW + c + 1] = t.y;
    }
    __syncthreads();

    // ---- S = (Q K^T) * 0.125, masked by adj, into LDS ----
    // wave -> (row tile r, column group g covering 256 columns)
    {
        const int r = w & 1;
        const int g = w >> 1;
        const int mrow = r * 16 + ln;

        // Hoist Q fragments: A(16x4 f32) lane l = row l%16, K = koff,koff+1
        v2f qf[16];
#pragma unroll
        for (int kk = 0; kk < 16; ++kk)
            qf[kk] = *(const v2f*)&sQ[mrow * QROW + kk * 4 + koff];

        for (int t = 0; t < 16; ++t) {
            const int col0 = g * 256 + t * 16;
            const int ncol = col0 + ln;
            const float* kp = &kb[(size_t)ncol * DHEAD + koff];

            v8f acc = {};
#pragma unroll
            for (int kk = 0; kk < 16; ++kk) {
                v2f bf = *(const v2f*)(kp + kk * 4);   // K[ncol, 4kk+koff..+1]
                acc = __builtin_amdgcn_wmma_f32_16x16x4_f32(
                    false, qf[kk], false, bf, (short)0, acc, false, false);
            }
            // C layout: VGPR j -> M = j + (l<16?0:8), N = l%16
#pragma unroll
            for (int j = 0; j < 8; ++j) {
                int m = j + ((l < 16) ? 0 : 8);
                int grow = row0 + r * 16 + m;
                int a_ = adjb[(size_t)grow * N_SEQ + col0 + ln];
                float val = (a_ > 0) ? acc[j] * 0.125f : NEGVAL;
                sP[(r * 16 + m) * SROW + col0 + ln] = val;
            }
        }
    }
    __syncthreads();

    // ---- Row softmax in LDS + non-temporal attn writeback ----
    // wave w handles rows 4w .. 4w+3
    for (int rr = 0; rr < 4; ++rr) {
        const int row = w * 4 + rr;
        float* srow = &sP[row * SROW];

        float mx = -INFINITY;
        for (int c = l; c < N_SEQ; c += 32) mx = fmaxf(mx, srow[c]);
#pragma unroll
        for (int off = 16; off > 0; off >>= 1)
            mx = fmaxf(mx, __shfl_xor(mx, off, 32));

        float sum = 0.0f;
        for (int c = l; c < N_SEQ; c += 32) {
            float e = __expf(srow[c] - mx);
            srow[c] = e;
            sum += e;
        }
#pragma unroll
        for (int off = 16; off > 0; off >>= 1)
            sum += __shfl_xor(sum, off, 32);

        const float inv = 1.0f / sum;
        float* arow = &attnb[(size_t)(row0 + row) * N_SEQ];
        for (int c = l; c < N_SEQ; c += 32) {
            float p = srow[c] * inv;
            srow[c] = p;                                  // keep P for P*V
            __builtin_nontemporal_store(p, &arow[c]);     // attn is write-only
        }
    }
    __syncthreads();

    // ---- O = P * V : each wave owns one 16x16 tile of the 32x64 O block ----
    {
        const int r2 = w >> 2;                 // row tile 0/1
        const int c2 = w & 3;                  // col tile 0..3
        const int m2 = r2 * 16 + ln;
        const float* srowp = &sP[m2 * SROW + koff];
        const float* vp = &vb[(size_t)koff * DHEAD + c2 * 16 + ln];

        v8f acc = {};
#pragma unroll 8
        for (int kk = 0; kk < N_SEQ; kk += 4) {
            v2f af = *(const v2f*)(srowp + kk);            // P[m2, kk+koff..+1]
            v2f bf;
            bf.x = vp[(size_t)kk * DHEAD];                 // V[kk+koff  ][n]
            bf.y = vp[(size_t)(kk + 1) * DHEAD];           // V[kk+koff+1][n]
            acc = __builtin_amdgcn_wmma_f32_16x16x4_f32(
                false, af, false, bf, (short)0, acc, false, false);
        }
#pragma unroll
        for (int j = 0; j < 8; ++j) {
            int m = j + ((l < 16) ? 0 : 8);
            __builtin_nontemporal_store(
                acc[j],
                &outb[(size_t)(row0 + r2 * 16 + m) * DHEAD + c2 * 16 + ln]);
        }
    }
}

extern "C" void kernel_launch(void* const* d_in, const int* in_sizes, int n_in,
                              void* d_out, int out_size, void* d_ws, size_t ws_size,
                              hipStream_t stream) {
    const float* q   = (const float*)d_in[0];
    const float* k   = (const float*)d_in[1];
    const float* v   = (const float*)d_in[2];
    const int*   adj = (const int*)d_in[3];

    float* out  = (float*)d_out;                              // [32,1024,64]
    float* attn = (float*)d_out + (size_t)32 * 1024 * 64;     // [32,1024,1024]

    const size_t shmem = (size_t)(BLKROWS * SROW + BLKROWS * QROW) * sizeof(float);
    hipFuncSetAttribute(
        reinterpret_cast<const void*>(GraphAtt_ScaledDotProduct_70471823392983_kernel),
        hipFuncAttributeMaxDynamicSharedMemorySize, (int)shmem);

    dim3 grid(N_SEQ / BLKROWS, 32);   // 32 row-blocks x 32 batches
    dim3 block(256);                  // 8 wave32s
    GraphAtt_ScaledDotProduct_70471823392983_kernel<<<grid, block, shmem, stream>>>(
        q, k, v, adj, out, attn);
}